// Hyper_GraphConv_56779467653538
// MI455X (gfx1250) — compile-verified
//
#include <hip/hip_runtime.h>

typedef __attribute__((ext_vector_type(2))) float v2f;
typedef __attribute__((ext_vector_type(8))) float v8f;

#define T_DIM   300
#define V_DIM   25
#define NEDGE   5
#define C_IN    3
#define K_DIM   15                     // NEDGE * C_IN (real K)
#define K_PAD   16                     // padded K for WMMA
#define P_DIM   (T_DIM * V_DIM)        // 7500 positions per batch
#define NT_TILES ((P_DIM + 15) / 16)   // 469 N-tiles of 16 columns
#define P_PAD   (NT_TILES * 16)        // 7504 padded positions
#define OUT_CH  64
#define N_B     64

// workspace layout (floats):
//   sup  : [N_B][K_PAD][P_PAD]   = 64*16*7504
//   wpad : [OUT_CH][K_PAD]       = 64*16   (column 15 zeroed)
#define SUP_FLOATS ((size_t)N_B * K_PAD * P_PAD)

// ---------------------------------------------------------------------------
// Prep: padded copy of w_mlp (64x15 -> 64x16, col 15 = 0)
// ---------------------------------------------------------------------------
__global__ __launch_bounds__(256) void prep_wpad(
    const float* __restrict__ w_mlp, float* __restrict__ wpad)
{
    int i = blockIdx.x * 256 + threadIdx.x;        // 4 blocks * 256 = 1024
    if (i < OUT_CH * K_PAD) {
        int o = i >> 4, k = i & 15;
        wpad[i] = (k < K_DIM) ? w_mlp[o * K_DIM + k] : 0.f;
    }
}

// ---------------------------------------------------------------------------
// Stage 1: build support[b, i=edge*3+c, p=t*25+v] (fp32, into workspace),
// plus zero K-pad plane i=15. One wave per (b,t); 8 waves / block.
// ---------------------------------------------------------------------------
__global__ __launch_bounds__(256) void stage1_support(
    const float* __restrict__ x,
    const float* __restrict__ h0, const float* __restrict__ h1,
    const float* __restrict__ h2, const float* __restrict__ h3,
    const float* __restrict__ h4,
    float* __restrict__ sup)
{
    __shared__ float sh_h[NEDGE][V_DIM * V_DIM];   // H[v*25+e] per edge type
    __shared__ float sh_de[NEDGE][V_DIM];          // 1/degree_e
    __shared__ float sh_w[8][32];
    __shared__ float sh_feat[8][C_IN][32];
    __shared__ float sh_dv[8][32];
    __shared__ float sh_g[8][C_IN][32];

    const float* hs[NEDGE] = {h0, h1, h2, h3, h4};
    const int tid = threadIdx.x;

    for (int i = tid; i < NEDGE * V_DIM * V_DIM; i += 256) {
        int et = i / (V_DIM * V_DIM);
        int r  = i - et * (V_DIM * V_DIM);
        sh_h[et][r] = hs[et][r];
    }
    __syncthreads();
    for (int i = tid; i < NEDGE * V_DIM; i += 256) {
        int et = i / V_DIM, e = i - et * V_DIM;
        float s = 0.f;
        for (int v = 0; v < V_DIM; ++v) s += sh_h[et][v * V_DIM + e];
        sh_de[et][e] = (s != 0.f) ? (1.f / s) : 0.f;
    }
    __syncthreads();

    const int wave = tid >> 5;
    const int lane = tid & 31;
    const int bt   = blockIdx.x * 8 + wave;       // 2400 blocks * 8 = 19200 exactly
    const int b    = bt / T_DIM;
    const int t    = bt - b * T_DIM;
    const bool av  = (lane < V_DIM);
    const int  li  = av ? lane : 0;               // clamped index for LDS reads
    const int chmap[NEDGE] = {6, 7, 9, 10, 11};

    // zero the K-pad plane (i = 15) for this (b,t)'s columns
    if (av)
        sup[((size_t)b * K_PAD + K_DIM) * P_PAD + (size_t)t * V_DIM + lane] = 0.f;

    for (int c = 0; c < C_IN; ++c) {
        float f = av ? x[((size_t)(b * 12 + c) * T_DIM + t) * V_DIM + lane] : 0.f;
        sh_feat[wave][c][lane] = f;
    }
    __syncthreads();

    for (int et = 0; et < NEDGE; ++et) {
        float wv = av ? x[((size_t)(b * 12 + chmap[et]) * T_DIM + t) * V_DIM + lane] : 0.f;
        sh_w[wave][lane] = wv;
        __syncthreads();

        // degree_v -> dv  (lane == v)
        float acc = 0.f;
        for (int e = 0; e < V_DIM; ++e)
            acc += sh_h[et][li * V_DIM + e] * sh_w[wave][e];
        float dvv = (av && acc > 0.f) ? rsqrtf(acc) : 0.f;
        sh_dv[wave][lane] = dvv;
        __syncthreads();

        // g[c,e] = de[e]*w[e] * sum_u H[u,e]*dv[u]*feat[c,u]   (lane == e)
        float g0 = 0.f, g1 = 0.f, g2 = 0.f;
        for (int u = 0; u < V_DIM; ++u) {
            float hd = sh_h[et][u * V_DIM + li] * sh_dv[wave][u];
            g0 += hd * sh_feat[wave][0][u];
            g1 += hd * sh_feat[wave][1][u];
            g2 += hd * sh_feat[wave][2][u];
        }
        float scale = av ? (sh_de[et][li] * sh_w[wave][li]) : 0.f;
        sh_g[wave][0][lane] = g0 * scale;
        sh_g[wave][1][lane] = g1 * scale;
        sh_g[wave][2][lane] = g2 * scale;
        __syncthreads();

        // sup[c,v] = dv[v] * sum_e H[v,e]*g[c,e]   (lane == v)
        float s0 = 0.f, s1 = 0.f, s2 = 0.f;
        for (int e = 0; e < V_DIM; ++e) {
            float hv = sh_h[et][li * V_DIM + e];
            s0 += hv * sh_g[wave][0][e];
            s1 += hv * sh_g[wave][1][e];
            s2 += hv * sh_g[wave][2][e];
        }
        if (av) {
            size_t base = ((size_t)b * K_PAD + et * C_IN) * P_PAD + (size_t)t * V_DIM + lane;
            sup[base            ] = s0 * dvv;
            sup[base +     P_PAD] = s1 * dvv;
            sup[base + 2 * P_PAD] = s2 * dvv;
        }
        __syncthreads();
    }
}

// ---------------------------------------------------------------------------
// Stage 2: out[b,o,p] = relu( sum_i wpad[o,i] * sup[b,i,p] + b_mlp[o] )
// fp32 WMMA 16x16x4: per wave one 64x16 output tile (4 M-tiles, B reused),
// K padded to 16 -> 4 chained WMMAs per accumulator, all loads unconditional.
// ---------------------------------------------------------------------------
__global__ __launch_bounds__(256) void stage2_mlp_wmma(
    const float* __restrict__ sup,
    const float* __restrict__ wpad,
    const float* __restrict__ b_mlp,
    float* __restrict__ out)
{
    const int wave = threadIdx.x >> 5;
    const int lane = threadIdx.x & 31;
    const int wid  = blockIdx.x * 8 + wave;        // 3752 blocks * 8 = 30016 = 64*469
    const int b    = wid / NT_TILES;
    const int nt   = wid - b * NT_TILES;

    const int col   = lane & 15;                   // N index
    const int khalf = (lane >> 4) << 1;            // 0 (lanes 0-15) or 2 (lanes 16-31)
    const int p     = nt * 16 + col;               // < P_PAD always

    const float* supb = sup + (size_t)b * K_PAD * P_PAD;

    // B regs for 4 K-steps: lane<16 -> {K0,K1}, lane>=16 -> {K2,K3}, column = lane%16
    v2f B[4];
#pragma unroll
    for (int kk = 0; kk < 4; ++kk) {
        int k0 = kk * 4 + khalf;
        B[kk].x = supb[(size_t)k0       * P_PAD + p];
        B[kk].y = supb[(size_t)(k0 + 1) * P_PAD + p];
    }

#pragma unroll
    for (int mt = 0; mt < 4; ++mt) {
        const int m = mt * 16 + col;               // A row for this lane
        v8f acc = {};
#pragma unroll
        for (int kk = 0; kk < 4; ++kk) {
            int k0 = kk * 4 + khalf;
            v2f A;
            A.x = wpad[m * K_PAD + k0    ];
            A.y = wpad[m * K_PAD + k0 + 1];
            // (neg_a, A, neg_b, B, c_mod, C, reuse_a, reuse_b)
            acc = __builtin_amdgcn_wmma_f32_16x16x4_f32(
                false, A, false, B[kk], (short)0, acc, false, false);
        }
        // D layout: reg j -> row mt*16 + j (+8 for lanes 16-31), col = lane%16
        const int rbase = mt * 16 + ((lane >> 4) << 3);
        if (p < P_DIM) {
#pragma unroll
            for (int j = 0; j < 8; ++j) {
                int o = rbase + j;
                float val = acc[j] + b_mlp[o];
                out[((size_t)b * OUT_CH + o) * P_DIM + p] = fmaxf(val, 0.f);
            }
        }
    }
}

// ---------------------------------------------------------------------------
extern "C" void kernel_launch(void* const* d_in, const int* in_sizes, int n_in,
                              void* d_out, int out_size, void* d_ws, size_t ws_size,
                              hipStream_t stream) {
    const float* x      = (const float*)d_in[0];
    const float* h0     = (const float*)d_in[1];
    const float* h1     = (const float*)d_in[2];
    const float* h2     = (const float*)d_in[3];
    const float* h3     = (const float*)d_in[4];
    const float* h4     = (const float*)d_in[5];
    const float* w_mlp  = (const float*)d_in[6];
    const float* b_mlp  = (const float*)d_in[7];
    float* out  = (float*)d_out;
    float* sup  = (float*)d_ws;                  // 64*16*7504 floats
    float* wpad = sup + SUP_FLOATS;              // 64*16 floats

    (void)in_sizes; (void)n_in; (void)out_size; (void)ws_size;

    prep_wpad<<<4, 256, 0, stream>>>(w_mlp, wpad);
    stage1_support<<<(N_B * T_DIM) / 8, 256, 0, stream>>>(x, h0, h1, h2, h3, h4, sup);
    stage2_mlp_wmma<<<(N_B * NT_TILES) / 8, 256, 0, stream>>>(sup, wpad, b_mlp, out);
}